// QuadraticFormSheafLearner_31842887533250
// MI455X (gfx1250) — compile-verified
//
#include <hip/hip_runtime.h>
#include <hip/hip_bf16.h>
#include <math.h>

typedef float v2f __attribute__((ext_vector_type(2)));
typedef float v8f __attribute__((ext_vector_type(8)));

#define IN_CH   32
#define KOUT    4          // prod(OUT_SHAPE) = 2*2
#define VWIDTH  (KOUT * IN_CH)   // 128 columns of the precomputed V matrix

// ---------------------------------------------------------------------------
// Phase 1: V[n, k*32 + j] = sum_i X[n,i] * T[k,i,j]
// Dense GEMM  X[50000 x 32]  @  W[32 x 128]   where W[i, k*32+j] = T[k,i,j].
// One wave computes one 16x16 tile of V via 8 chained V_WMMA_F32_16X16X4_F32.
// Block = 256 threads = 8 waves -> 8 column tiles (covers all 128 cols),
// grid.x = ceil(n_nodes/16) row tiles.
//
// Fragment layouts per CDNA5 ISA 7.12.2 (wave32):
//   A 16x4 f32 : lane m=lane&15 is row M; lanes 0-15 hold K=k0,k0+1 in v[0],v[1],
//                lanes 16-31 hold K=k0+2,k0+3.
//   B 4x16 f32 : lane n=lane&15 is column N; same K split as A (mirrored).
//   C/D 16x16  : VGPR r holds row M=r (lanes 0-15) and M=r+8 (lanes 16-31).
// ---------------------------------------------------------------------------
__global__ void __launch_bounds__(256)
sheaf_precompute_wmma(const float* __restrict__ X,
                      const float* __restrict__ T,
                      float* __restrict__ V,
                      int n_nodes) {
    const int lane = threadIdx.x & 31;
    const int wave = threadIdx.x >> 5;          // 0..7 -> column tile
    const int row0 = blockIdx.x * 16;

    const int m  = lane & 15;                   // row within A tile / col within B,D tile
    const int hi = lane >> 4;                   // lane half selects K sub-pair

    int arow = row0 + m;                        // clamp OOB reads (stores are guarded)
    if (arow >= n_nodes) arow = n_nodes - 1;
    const float* __restrict__ Xrow = X + (size_t)arow * IN_CH;

    const int col  = wave * 16 + m;             // global output column 0..127
    const int kmat = col >> 5;                  // which T[k]
    const int jj   = col & 31;                  // column within T[k]
    // W[i, col] = T[kmat*1024 + i*32 + jj]  (stride 32 in i)
    const float* __restrict__ Wcol = T + kmat * (IN_CH * IN_CH) + jj;

    v8f acc = {};
#pragma unroll
    for (int k0 = 0; k0 < IN_CH; k0 += 4) {
        const int ka = k0 + hi * 2;
        v2f a, b;
        a.x = Xrow[ka];
        a.y = Xrow[ka + 1];
        b.x = Wcol[ka * IN_CH];
        b.y = Wcol[(ka + 1) * IN_CH];
        // (neg_a, A, neg_b, B, c_mod, C, reuse_a, reuse_b)
        acc = __builtin_amdgcn_wmma_f32_16x16x4_f32(
            false, a, false, b, (short)0, acc, false, false);
    }

#pragma unroll
    for (int r = 0; r < 8; ++r) {
        const int row = row0 + r + hi * 8;
        if (row < n_nodes) V[(size_t)row * VWIDTH + col] = acc[r];
    }
}

// ---------------------------------------------------------------------------
// Phase 2 (memory-bound): per edge e,
//   maps[e,k] = dot(V[src[e], k*32 .. k*32+31], x[dst[e], :]);  out = tanh(maps)
// One thread per edge. Gathers hit the 192MB L2 (x:6.4MB + V:25.6MB resident);
// HBM traffic is dominated by streaming edge_index in and the output out.
// ---------------------------------------------------------------------------
__global__ void __launch_bounds__(256)
sheaf_edge_kernel(const float* __restrict__ V,
                  const float* __restrict__ X,
                  const long long* __restrict__ edge_index,
                  float* __restrict__ out,
                  int n_edges) {
    const int e = blockIdx.x * blockDim.x + threadIdx.x;
    if (e >= n_edges) return;

    const long long s = edge_index[e];                       // src row
    const long long d = edge_index[(long long)n_edges + e];  // dst row

    // x[dst] : 32 floats as 8 x float4 (global_load_b128)
    const float4* __restrict__ xd = (const float4*)(X + (size_t)d * IN_CH);
    float4 xv[8];
#pragma unroll
    for (int q = 0; q < 8; ++q) xv[q] = xd[q];

    const float4* __restrict__ vs = (const float4*)(V + (size_t)s * VWIDTH);

    float4 res;
    float* rp = (float*)&res;
#pragma unroll
    for (int k = 0; k < KOUT; ++k) {
        float acc = 0.0f;
#pragma unroll
        for (int q = 0; q < 8; ++q) {
            const float4 vv = vs[k * 8 + q];
            acc = fmaf(vv.x, xv[q].x, acc);
            acc = fmaf(vv.y, xv[q].y, acc);
            acc = fmaf(vv.z, xv[q].z, acc);
            acc = fmaf(vv.w, xv[q].w, acc);
        }
        rp[k] = tanhf(acc);
    }
    ((float4*)out)[e] = res;   // 16B store per edge, fully coalesced
}

// ---------------------------------------------------------------------------
// Fallback (only if d_ws is too small for V): direct per-edge quadratic form
// with T staged in LDS (16 KB). Deterministic given fixed ws_size.
// ---------------------------------------------------------------------------
__global__ void __launch_bounds__(256)
sheaf_direct_kernel(const float* __restrict__ X,
                    const long long* __restrict__ edge_index,
                    const float* __restrict__ T,
                    float* __restrict__ out,
                    int n_edges) {
    __shared__ float sT[KOUT * IN_CH * IN_CH];
    for (int i = threadIdx.x; i < KOUT * IN_CH * IN_CH; i += blockDim.x)
        sT[i] = T[i];
    __syncthreads();

    const int e = blockIdx.x * blockDim.x + threadIdx.x;
    if (e >= n_edges) return;

    const long long s = edge_index[e];
    const long long d = edge_index[(long long)n_edges + e];

    float xs[IN_CH], xd[IN_CH];
    const float4* ps = (const float4*)(X + (size_t)s * IN_CH);
    const float4* pd = (const float4*)(X + (size_t)d * IN_CH);
#pragma unroll
    for (int q = 0; q < 8; ++q) {
        ((float4*)xs)[q] = ps[q];
        ((float4*)xd)[q] = pd[q];
    }

    float4 res;
    float* rp = (float*)&res;
#pragma unroll
    for (int k = 0; k < KOUT; ++k) {
        float acc = 0.0f;
        const float* tk = sT + k * IN_CH * IN_CH;
        for (int i = 0; i < IN_CH; ++i) {
            float r = 0.0f;
#pragma unroll
            for (int j = 0; j < IN_CH; ++j)
                r = fmaf(tk[i * IN_CH + j], xd[j], r);
            acc = fmaf(xs[i], r, acc);
        }
        rp[k] = tanhf(acc);
    }
    ((float4*)out)[e] = res;
}

extern "C" void kernel_launch(void* const* d_in, const int* in_sizes, int n_in,
                              void* d_out, int out_size, void* d_ws, size_t ws_size,
                              hipStream_t stream) {
    const float*     x   = (const float*)d_in[0];        // [N, 32] fp32
    const long long* ei  = (const long long*)d_in[1];    // [2, E] int64
    const float*     ten = (const float*)d_in[2];        // [4, 32, 32] fp32
    float*           out = (float*)d_out;                // [E, 4] fp32 (tanh'd)

    const int n_nodes = in_sizes[0] / IN_CH;
    const int n_edges = in_sizes[1] / 2;

    const size_t needV = (size_t)n_nodes * VWIDTH * sizeof(float);

    if (ws_size >= needV) {
        float* V = (float*)d_ws;
        const int row_tiles = (n_nodes + 15) / 16;
        sheaf_precompute_wmma<<<row_tiles, 256, 0, stream>>>(x, ten, V, n_nodes);

        const int blocks = (n_edges + 255) / 256;
        sheaf_edge_kernel<<<blocks, 256, 0, stream>>>(V, x, ei, out, n_edges);
    } else {
        const int blocks = (n_edges + 255) / 256;
        sheaf_direct_kernel<<<blocks, 256, 0, stream>>>(x, ei, ten, out, n_edges);
    }
}